// Cheb_NET_51977694216540
// MI455X (gfx1250) — compile-verified
//
#include <hip/hip_runtime.h>
#include <hip/hip_bf16.h>
#include <math.h>

typedef __attribute__((ext_vector_type(2))) float v2f;
typedef __attribute__((ext_vector_type(8))) float v8f;

#define A_STRIDE 66   // 64 + 2 pad: float2-aligned, kills 16-way bank conflicts on A reads
#define W_STRIDE 68   // banks (4n + k0) % 64 -> conflict-free B-fragment b64 reads

// ---------------------------------------------------------------------------
// 1) degree / normalization precompute
// ---------------------------------------------------------------------------
__global__ void k_zero(float* __restrict__ deg, unsigned* __restrict__ maxbits, int N) {
    int i = blockIdx.x * blockDim.x + threadIdx.x;
    if (i < N) deg[i] = 0.0f;
    if (i == 0) *maxbits = 0u;
}

__global__ void k_count_deg(const int* __restrict__ src, float* __restrict__ deg, int E) {
    int e = blockIdx.x * blockDim.x + threadIdx.x;
    if (e < E) atomicAdd(&deg[src[e]], 1.0f);
}

__global__ void k_max_deg(const float* __restrict__ deg, unsigned* __restrict__ maxbits, int N) {
    __shared__ float red[256];
    int i = blockIdx.x * blockDim.x + threadIdx.x;
    float v = (i < N) ? deg[i] : 0.0f;
    red[threadIdx.x] = v;
    __syncthreads();
    for (int off = 128; off > 0; off >>= 1) {
        if (threadIdx.x < off) red[threadIdx.x] = fmaxf(red[threadIdx.x], red[threadIdx.x + off]);
        __syncthreads();
    }
    // deg >= 0, so IEEE bit pattern compares correctly as unsigned
    if (threadIdx.x == 0) atomicMax(maxbits, __float_as_uint(red[0]));
}

__device__ __forceinline__ float cheb_scale(const unsigned* maxbits) {
    float md  = __uint_as_float(*maxbits);
    float lam = 2.0f * fmaxf(md, -1.0f);   // jnp.maximum(max(deg), max(-w)) with w==1
    return 2.0f / lam;
}

__global__ void k_norm_loop(const float* __restrict__ deg, const unsigned* __restrict__ maxbits,
                            float* __restrict__ nloop, int N) {
    int i = blockIdx.x * blockDim.x + threadIdx.x;
    if (i >= N) return;
    float s = cheb_scale(maxbits);
    nloop[i] = deg[i] * s - 1.0f;
}

// ---------------------------------------------------------------------------
// 2) SpMM:  t[i] = norm_loop[i]*h[i]  then  t[dst] += (-scale) * h[src]
// ---------------------------------------------------------------------------
__global__ void k_spmm_init(const float* __restrict__ h, const float* __restrict__ nloop,
                            float* __restrict__ t, int total /* N*64 */) {
    int i = blockIdx.x * blockDim.x + threadIdx.x;
    if (i < total) t[i] = nloop[i >> 6] * h[i];
}

__global__ void k_spmm_edge(const float* __restrict__ h, const int* __restrict__ src,
                            const int* __restrict__ dst, const unsigned* __restrict__ maxbits,
                            float* __restrict__ t, int E) {
    int tid = blockIdx.x * blockDim.x + threadIdx.x;
    if (tid >= E * 16) return;
    float nscale = -cheb_scale(maxbits);
    int e = tid >> 4;
    int c = (tid & 15) << 2;                 // 4-float chunk
    int s = src[e];
    int d = dst[e];
    const float4 v = *reinterpret_cast<const float4*>(h + (size_t)s * 64 + c);
    float* o = t + (size_t)d * 64 + c;
    atomicAdd(o + 0, nscale * v.x);
    atomicAdd(o + 1, nscale * v.y);
    atomicAdd(o + 2, nscale * v.z);
    atomicAdd(o + 3, nscale * v.w);
}

// ---------------------------------------------------------------------------
// 3) Fused dual GEMM:  out = H@W0 + T@W1 + b  (+ optional ReLU)
//    K = 64 always.  128-thread blocks (4 waves), persistent over 16-row tiles.
//    Weights staged once per block into LDS, transposed + zero-padded, so the
//    wmma B fragment {W[k0][n], W[k0+1][n]} is one aligned ds_load_b64 and the
//    register footprint stays small (full occupancy).
//    Wave w owns output columns [16w, 16w+16); v_wmma_f32_16x16x4_f32.
// ---------------------------------------------------------------------------
__global__ __launch_bounds__(128)
void k_cheb_gemm(const float* __restrict__ H, const float* __restrict__ T,
                 const float* __restrict__ W0, const float* __restrict__ W1,
                 const float* __restrict__ bias, float* __restrict__ out,
                 int Nnodes, int Cout, int doRelu) {
    __shared__ float As[16 * A_STRIDE];
    __shared__ float Ts[16 * A_STRIDE];
    __shared__ float Wt0[64 * W_STRIDE];
    __shared__ float Wt1[64 * W_STRIDE];

    const int tid = threadIdx.x;

    // one-time weight staging: Wt[n][k] = W[k][n], zero-padded for n >= Cout
    for (int i = tid; i < 4096; i += 128) {
        int k = i >> 6;            // 0..63 (fan_in)
        int n = i & 63;            // 0..63 (padded fan_out), global read coalesced over n
        float w0 = (n < Cout) ? W0[k * Cout + n] : 0.0f;
        float w1 = (n < Cout) ? W1[k * Cout + n] : 0.0f;
        Wt0[n * W_STRIDE + k] = w0;
        Wt1[n * W_STRIDE + k] = w1;
    }

    const int lane = tid & 31;
    const int wv   = tid >> 5;
    const int hf   = lane >> 4;              // K-half selector (A/B layout)
    const int lcol = lane & 15;
    const int n    = wv * 16 + lcol;         // output column
    const int kofs = 2 * hf;

    const int nTiles = (Nnodes + 15) >> 4;
    for (int tile = blockIdx.x; tile < nTiles; tile += gridDim.x) {
        const int row0 = tile * 16;
        __syncthreads();                     // previous iteration done with LDS

        // stage 16x64 H and T tiles into LDS, coalesced
        const size_t base = (size_t)row0 * 64;
        #pragma unroll
        for (int it = 0; it < 8; ++it) {
            int j = tid + it * 128;          // 0..1023
            int r = j >> 6, c = j & 63;
            float hv = 0.0f, tv = 0.0f;
            if (row0 + r < Nnodes) { hv = H[base + j]; tv = T[base + j]; }
            As[r * A_STRIDE + c] = hv;
            Ts[r * A_STRIDE + c] = tv;
        }
        __syncthreads();

        v8f acc0 = {};                       // H @ W0
        v8f acc1 = {};                       // T @ W1  (independent chain -> XDL overlap)
        #pragma unroll
        for (int k = 0; k < 16; ++k) {
            int k0 = 4 * k + kofs;
            v2f a0 = *(const v2f*)&As[lcol * A_STRIDE + k0];
            v2f b0 = *(const v2f*)&Wt0[n * W_STRIDE + k0];
            acc0 = __builtin_amdgcn_wmma_f32_16x16x4_f32(false, a0, false, b0,
                                                         (short)0, acc0, false, false);
            v2f a1 = *(const v2f*)&Ts[lcol * A_STRIDE + k0];
            v2f b1 = *(const v2f*)&Wt1[n * W_STRIDE + k0];
            acc1 = __builtin_amdgcn_wmma_f32_16x16x4_f32(false, a1, false, b1,
                                                         (short)0, acc1, false, false);
        }

        if (n < Cout) {
            float bv = bias[n];
            #pragma unroll
            for (int v = 0; v < 8; ++v) {
                int r = row0 + v + 8 * hf;   // D layout: lanes 16-31 hold M = v+8
                if (r < Nnodes) {
                    float val = acc0[v] + acc1[v] + bv;
                    if (doRelu) val = fmaxf(val, 0.0f);
                    out[(size_t)r * Cout + n] = val;
                }
            }
        }
    }
}

// ---------------------------------------------------------------------------
// 4) log_softmax over 40 columns, one wave32 per row
// ---------------------------------------------------------------------------
__global__ __launch_bounds__(128)
void k_logsoftmax40(float* __restrict__ z, int N) {
    int wv   = threadIdx.x >> 5;
    int lane = threadIdx.x & 31;
    int row  = blockIdx.x * 4 + wv;
    if (row >= N) return;
    float* p = z + (size_t)row * 40;
    float v0 = p[lane];                                  // lane 0..31 < 40 always valid
    float v1 = (lane < 8) ? p[32 + lane] : -3.0e38f;
    float m = fmaxf(v0, v1);
    #pragma unroll
    for (int off = 16; off > 0; off >>= 1) m = fmaxf(m, __shfl_xor(m, off, 32));
    float s = expf(v0 - m) + ((lane < 8) ? expf(v1 - m) : 0.0f);
    #pragma unroll
    for (int off = 16; off > 0; off >>= 1) s += __shfl_xor(s, off, 32);
    float lse = m + logf(s);
    p[lane] = v0 - lse;
    if (lane < 8) p[32 + lane] = v1 - lse;
}

// ---------------------------------------------------------------------------
// launcher
// ---------------------------------------------------------------------------
extern "C" void kernel_launch(void* const* d_in, const int* in_sizes, int n_in,
                              void* d_out, int out_size, void* d_ws, size_t ws_size,
                              hipStream_t stream) {
    const float* x   = (const float*)d_in[0];
    const int*   ei  = (const int*)d_in[1];
    const float* W00 = (const float*)d_in[2];
    const float* W10 = (const float*)d_in[3];
    const float* b0  = (const float*)d_in[4];
    const float* W01 = (const float*)d_in[5];
    const float* W11 = (const float*)d_in[6];
    const float* b1  = (const float*)d_in[7];
    const float* W02 = (const float*)d_in[8];
    const float* W12 = (const float*)d_in[9];
    const float* b2  = (const float*)d_in[10];

    const int N = in_sizes[0] / 64;
    const int E = in_sizes[1] / 2;
    const int* src = ei;
    const int* dst = ei + E;

    float*    wsf     = (float*)d_ws;
    float*    deg     = wsf;
    float*    nloop   = wsf + (size_t)N;
    unsigned* maxbits = (unsigned*)(wsf + 2 * (size_t)N);
    float*    bufH    = wsf + 2 * (size_t)N + 16;
    float*    bufT    = bufH + (size_t)N * 64;

    const int TPB = 256;
    const int gN   = (N + TPB - 1) / TPB;
    const int gE   = (E + TPB - 1) / TPB;
    const int gNC  = (N * 64 + TPB - 1) / TPB;
    const int gE16 = (E * 16 + TPB - 1) / TPB;
    const int nTiles = (N + 15) / 16;
    const int gTile  = (nTiles < 2048) ? nTiles : 2048;   // persistent blocks

    // normalization precompute
    k_zero<<<gN, TPB, 0, stream>>>(deg, maxbits, N);
    k_count_deg<<<gE, TPB, 0, stream>>>(src, deg, E);
    k_max_deg<<<gN, TPB, 0, stream>>>(deg, maxbits, N);
    k_norm_loop<<<gN, TPB, 0, stream>>>(deg, maxbits, nloop, N);

    // layer 0: h = x -> bufH (ReLU)
    k_spmm_init<<<gNC, TPB, 0, stream>>>(x, nloop, bufT, N * 64);
    k_spmm_edge<<<gE16, TPB, 0, stream>>>(x, src, dst, maxbits, bufT, E);
    k_cheb_gemm<<<gTile, 128, 0, stream>>>(x, bufT, W00, W10, b0, bufH, N, 64, 1);

    // layer 1: bufH -> bufH (in-place safe: rows staged to LDS before store) (ReLU)
    k_spmm_init<<<gNC, TPB, 0, stream>>>(bufH, nloop, bufT, N * 64);
    k_spmm_edge<<<gE16, TPB, 0, stream>>>(bufH, src, dst, maxbits, bufT, E);
    k_cheb_gemm<<<gTile, 128, 0, stream>>>(bufH, bufT, W01, W11, b1, bufH, N, 64, 1);

    // layer 2: bufH -> d_out (40 cols, no ReLU), then log_softmax in place
    k_spmm_init<<<gNC, TPB, 0, stream>>>(bufH, nloop, bufT, N * 64);
    k_spmm_edge<<<gE16, TPB, 0, stream>>>(bufH, src, dst, maxbits, bufT, E);
    k_cheb_gemm<<<gTile, 128, 0, stream>>>(bufH, bufT, W02, W12, b2, (float*)d_out, N, 40, 0);

    k_logsoftmax40<<<(N + 3) / 4, 128, 0, stream>>>((float*)d_out, N);
}